// VectorQuantizer_55035710931023
// MI455X (gfx1250) — compile-verified
//
#include <hip/hip_runtime.h>

typedef __attribute__((ext_vector_type(2))) float v2f;
typedef __attribute__((ext_vector_type(8))) float v8f;
typedef int v4i __attribute__((vector_size(4 * sizeof(int))));
typedef __attribute__((address_space(1))) v4i global_v4i;
typedef __attribute__((address_space(3))) v4i lds_v4i;

#define KCODES 1024
#define DDIM   256
#define NVEC   32768       // 32*32*32 rows
#define HW     1024        // 32*32
#define BSTRIDE (DDIM*HW)  // 262144 floats per batch image

// d_out layout (floats): [0]=loss, [1..8388608]=quantized, [8388609]=perplexity,
// [8388610..]=encodings (N*K)
#define QOFF 1
#define POFF 8388609
#define EOFF 8388610

// workspace layout (bytes)
#define WS_ENORM    0        // 1024 floats
#define WS_IDX      4096     // 32768 ints
#define WS_COUNTS   135168   // 1024 ints
#define WS_PARTIAL  139264   // 8192 floats

// LDS B-tile: 16 code rows x 256 d, row-major, padded pitch for bank spread
#define BPITCH 260                 // floats per code row (260 % 64 == 4)
#define BUFSZ  (16 * BPITCH)       // floats per buffer (16640 B)

#if __has_builtin(__builtin_amdgcn_global_load_async_to_lds_b128)
#define HAVE_ASYNC_LDS 1
#else
#define HAVE_ASYNC_LDS 0
#endif

#if __has_builtin(__builtin_amdgcn_s_wait_asynccnt)
#define WAIT_ASYNC(n) __builtin_amdgcn_s_wait_asynccnt(n)
#else
#define WAIT_ASYNC(n) asm volatile("s_wait_asynccnt " #n)
#endif

// ---------------------------------------------------------------------------
// Kernel 1: codebook norms ||e_k||^2 + zero counters. One wave per code row.
// ---------------------------------------------------------------------------
__global__ __launch_bounds__(128) void vq_prep(const float* __restrict__ emb,
                                               float* __restrict__ enorm,
                                               int* __restrict__ counts) {
  int tid  = threadIdx.x;
  int lane = tid & 31;
  int row  = blockIdx.x * 4 + (tid >> 5);
  float s = 0.f;
  #pragma unroll
  for (int j = 0; j < 8; ++j) {
    float v = emb[row * DDIM + lane + j * 32];
    s += v * v;
  }
  #pragma unroll
  for (int off = 16; off >= 1; off >>= 1) s += __shfl_xor(s, off, 32);
  if (lane == 0) enorm[row] = s;
  if (blockIdx.x < 8) counts[blockIdx.x * 128 + tid] = 0;
}

// ---------------------------------------------------------------------------
// Stage one 16-code x 256-d tile (16 KB) into LDS buffer `buf`.
// 1024 contiguous 16B chunks; thread t takes chunks t, t+128, ... (8 each) ->
// perfectly coalesced on the global side. Async path uses
// GLOBAL_LOAD_ASYNC_TO_LDS_B128 (ASYNCcnt) to overlap with WMMA compute.
// ---------------------------------------------------------------------------
__device__ __forceinline__ void stage_tile(const float* __restrict__ emb,
                                           float* __restrict__ ldsB,
                                           int kb, int buf, int tid) {
  #pragma unroll
  for (int j = 0; j < 8; ++j) {
    int cc = j * 128 + tid;          // chunk id 0..1023
    int c  = cc >> 6;                // code row 0..15
    int jc = cc & 63;                // 16B chunk within row
    const float* g = emb + (kb * 16 + c) * DDIM + jc * 4;
    float*       l = ldsB + buf * BUFSZ + c * BPITCH + jc * 4;
#if HAVE_ASYNC_LDS
    __builtin_amdgcn_global_load_async_to_lds_b128(
        (global_v4i*)(void*)g, (lds_v4i*)l, 0, 0);
#else
    *(float4*)l = *(const float4*)g;
#endif
  }
}

// ---------------------------------------------------------------------------
// Kernel 2: argmin_k ( ||e_k||^2 - 2 x.e_k ) via fp32 WMMA.
// 128 threads = 4 waves; each wave owns a 16-row tile (block: 64 rows).
// A tile (16x256 fp32) lives in VGPRs as 64 v2f fragments (ISA 7.12.2 layout).
// B tile double-buffered in LDS; each fragment is one aligned ds_load_b64,
// conflict-free (banks (4c+d0) mod 64 cover all 64 banks once per access).
// ---------------------------------------------------------------------------
__global__ __launch_bounds__(128) void vq_argmin(const float* __restrict__ latents,
                                                 const float* __restrict__ emb,
                                                 const float* __restrict__ enorm,
                                                 int* __restrict__ outIdx) {
  __shared__ __align__(16) float ldsB[2 * BUFSZ];

  int tid  = threadIdx.x;
  int lane = tid & 31;
  int wave = tid >> 5;
  int ksel = (lane < 16) ? 0 : 2;   // A/B: lanes 0-15 hold K=0,1; 16-31 K=2,3
  int code = lane & 15;             // column within 16-code chunk
  int rowBase = blockIdx.x * 64 + wave * 16;

  // --- A fragments: row rowBase+code, strided over D (transpose gather,
  // coalesced across lanes: 16 consecutive rows share b,h and step w) ---
  int rowA = rowBase + code;
  const float* xb = latents + (rowA >> 10) * BSTRIDE + (rowA & 1023);
  v2f afrag[64];
  #pragma unroll
  for (int dc = 0; dc < 64; ++dc) {
    int d0 = dc * 4 + ksel;
    afrag[dc].x = xb[d0 * HW];
    afrag[dc].y = xb[(d0 + 1) * HW];
  }

  float minv[8];
  int   mini[8];
  #pragma unroll
  for (int r = 0; r < 8; ++r) { minv[r] = 3.4e38f; mini[r] = 0; }

  stage_tile(emb, ldsB, 0, 0, tid);   // prologue: tile 0 -> buffer 0

  for (int kb = 0; kb < 64; ++kb) {
    int cur = kb & 1;
    if (kb + 1 < 64) stage_tile(emb, ldsB, kb + 1, cur ^ 1, tid);
#if HAVE_ASYNC_LDS
    // drain this tile's 8 async copies; leave next tile's 8 in flight
    if (kb + 1 < 64) WAIT_ASYNC(8); else WAIT_ASYNC(0);
#endif
    __syncthreads();                  // tile `cur` visible block-wide

    const float* bbase = ldsB + cur * BUFSZ + code * BPITCH + ksel;
    v8f acc = {0.f, 0.f, 0.f, 0.f, 0.f, 0.f, 0.f, 0.f};
    #pragma unroll
    for (int dc = 0; dc < 64; ++dc) {
      v2f b = *(const v2f*)(bbase + dc * 4);   // one ds_load_b64
      acc = __builtin_amdgcn_wmma_f32_16x16x4_f32(false, afrag[dc], false, b,
                                                  (short)0, acc, false, false);
    }

    float en = enorm[kb * 16 + code];
    #pragma unroll
    for (int r = 0; r < 8; ++r) {
      float dval = en - 2.0f * acc[r];  // + ||x||^2 is row-constant: irrelevant
      if (dval < minv[r]) { minv[r] = dval; mini[r] = kb * 16 + code; }
    }
    __syncthreads();                  // protect buffer cur until reads done
  }

  // reduce across the 16 lanes holding the same output row (C layout: VGPR r ->
  // row r (lanes 0-15) / row r+8 (lanes 16-31); lane&15 -> column)
  #pragma unroll
  for (int r = 0; r < 8; ++r) {
    float v = minv[r]; int id = mini[r];
    #pragma unroll
    for (int off = 8; off >= 1; off >>= 1) {
      float v2 = __shfl_xor(v, off, 32);
      int   i2 = __shfl_xor(id, off, 32);
      if (v2 < v || (v2 == v && i2 < id)) { v = v2; id = i2; }
    }
    if (code == 0) {
      int row = rowBase + r + ((lane < 16) ? 0 : 8);
      outIdx[row] = id;
    }
  }
}

// ---------------------------------------------------------------------------
// Kernel 3: quantized output (already [B,D,H,W] layout), loss partials, counts.
// One block per (b,d); threads sweep hw -> fully coalesced loads/stores.
// ---------------------------------------------------------------------------
__global__ __launch_bounds__(256) void vq_gather(const float* __restrict__ latents,
                                                 const float* __restrict__ emb,
                                                 const int* __restrict__ idx,
                                                 float* __restrict__ quant,
                                                 float* __restrict__ partial,
                                                 int* __restrict__ counts) {
  __shared__ float red[256];
  int b = blockIdx.x >> 8;
  int d = blockIdx.x & 255;
  int tid = threadIdx.x;
  int base = b * BSTRIDE + d * HW;
  float s = 0.f;
  #pragma unroll
  for (int j = 0; j < 4; ++j) {
    int hw = tid + j * 256;
    int n  = b * HW + hw;
    int id = idx[n];
    float q  = emb[id * DDIM + d];
    float xv = latents[base + hw];
    float df = q - xv;
    s += df * df;
    quant[base + hw] = q;
    if (d == 0) atomicAdd(&counts[id], 1);   // integer: deterministic
  }
  red[tid] = s;
  __syncthreads();
  for (int off = 128; off >= 1; off >>= 1) {
    if (tid < off) red[tid] += red[tid + off];
    __syncthreads();
  }
  if (tid == 0) partial[blockIdx.x] = red[0];
}

// ---------------------------------------------------------------------------
// Kernel 4: one-hot encodings [N,K]. Output base is only 8B aligned -> float2.
// ---------------------------------------------------------------------------
__global__ __launch_bounds__(256) void vq_encodings(const int* __restrict__ idx,
                                                    float* __restrict__ enc) {
  int n  = blockIdx.x;
  int id = idx[n];                        // uniform within block (scalarized)
  float2* dst = (float2*)(enc + (size_t)n * KCODES);
  int tid = threadIdx.x;
  #pragma unroll
  for (int j = 0; j < 2; ++j) {
    int k0 = (j * 256 + tid) * 2;
    float2 v;
    v.x = (k0     == id) ? 1.f : 0.f;
    v.y = (k0 + 1 == id) ? 1.f : 0.f;
    dst[j * 256 + tid] = v;
  }
}

// ---------------------------------------------------------------------------
// Kernel 5: deterministic final reductions -> loss + perplexity.
// ---------------------------------------------------------------------------
__global__ __launch_bounds__(256) void vq_finalize(const float* __restrict__ partial,
                                                   const int* __restrict__ counts,
                                                   float* __restrict__ out) {
  __shared__ float red[256];
  int tid = threadIdx.x;
  float s = 0.f;
  for (int i = tid; i < 8192; i += 256) s += partial[i];
  red[tid] = s;
  __syncthreads();
  for (int off = 128; off >= 1; off >>= 1) {
    if (tid < off) red[tid] += red[tid + off];
    __syncthreads();
  }
  float lossSum = red[0];
  __syncthreads();

  float h = 0.f;
  for (int k = tid; k < KCODES; k += 256) {
    float p = (float)counts[k] * (1.0f / (float)NVEC);
    h += p * logf(p + 1e-10f);
  }
  red[tid] = h;
  __syncthreads();
  for (int off = 128; off >= 1; off >>= 1) {
    if (tid < off) red[tid] += red[tid + off];
    __syncthreads();
  }
  if (tid == 0) {
    // q_loss + beta*e_loss == 1.25 * mean((q-x)^2) in the forward pass
    out[0]    = 1.25f * lossSum / (float)((long)NVEC * DDIM);
    out[POFF] = expf(-red[0]);
  }
}

extern "C" void kernel_launch(void* const* d_in, const int* in_sizes, int n_in,
                              void* d_out, int out_size, void* d_ws, size_t ws_size,
                              hipStream_t stream) {
  const float* latents = (const float*)d_in[0];
  const float* emb     = (const float*)d_in[1];
  float* out = (float*)d_out;
  char*  ws  = (char*)d_ws;

  float* enorm   = (float*)(ws + WS_ENORM);
  int*   idx     = (int*)(ws + WS_IDX);
  int*   counts  = (int*)(ws + WS_COUNTS);
  float* partial = (float*)(ws + WS_PARTIAL);

  float* quant = out + QOFF;
  float* enc   = out + EOFF;

  vq_prep     <<<256,   128, 0, stream>>>(emb, enorm, counts);
  vq_argmin   <<<512,   128, 0, stream>>>(latents, emb, enorm, idx);
  vq_gather   <<<8192,  256, 0, stream>>>(latents, emb, idx, quant, partial, counts);
  vq_encodings<<<NVEC,  256, 0, stream>>>(idx, enc);
  vq_finalize <<<1,     256, 0, stream>>>(partial, counts, out);
}